// gen_Cinv_rows_51556787421858
// MI455X (gfx1250) — compile-verified
//
#include <hip/hip_runtime.h>

// CDNA5 / gfx1250, wave32. One workgroup (256 thr = 8 waves) per output block.
// Pipeline per block: scatter-build C_sub (64x64, LDS) -> gram = C*C^T via
// V_WMMA_F32_16X16X4_F32 -> Gauss-Jordan inverse of gram in LDS (SPD, no
// pivoting) -> full = inv * C_sub[:,16:48] via WMMA -> transposed store.

typedef float v2f __attribute__((ext_vector_type(2)));
typedef float v8f __attribute__((ext_vector_type(8)));

#define NN     512
#define NI     64
#define NI_HALF 32
#define KROW   32
#define NNZ    (NN * NN * 13)       // 3407872
#define CS     65                   // C_sub LDS row stride (pad vs 64 banks)
#define AS     130                  // augmented [gram | I] row stride
#define FS     33                   // full_sub (64x32) row stride

__global__ __launch_bounds__(256) void gen_cinv_rows_kernel(
    const float* __restrict__ Cs,
    const long long* __restrict__ coo_r,
    const long long* __restrict__ coo_c,
    float* __restrict__ out)
{
    __shared__ float Csub[NI * CS];   // 16.6 KB
    __shared__ float aug [NI * AS];   // 33.3 KB : [gram | inv]
    __shared__ float fs  [NI * FS];   //  8.4 KB : inv * C_sub[:,16:48]

    const int tid = threadIdx.x;
    const int b   = blockIdx.x;
    const int i   = KROW / 2 + KROW * b;      // window center row

    // ---- 1) zero C_sub ----------------------------------------------------
    for (int idx = tid; idx < NI * CS; idx += 256) Csub[idx] = 0.0f;
    __syncthreads();

    // ---- 2) scatter 64*13 band entries into the window (ds_add_f32) -------
    const long long base = (long long)(i - NI_HALF) * 13;
    for (int e = tid; e < NI * 13; e += 256) {
        long long g   = base + e;
        bool valid    = (g >= 0) && (g < (long long)NNZ);
        long long gc  = g < 0 ? 0 : (g >= NNZ ? (long long)NNZ - 1 : g);
        float v       = valid ? Cs[gc] : 0.0f;
        int r         = (int)coo_r[gc] - i + NI_HALF;
        int c         = (int)coo_c[gc] - i + NI_HALF;
        if (valid && r >= 0 && r < NI && c >= 0 && c < NI && v != 0.0f)
            atomicAdd(&Csub[r * CS + c], v);
    }
    __syncthreads();

    // ---- 3) boundary diagonal fix-ups (incl. the d==0 right-edge quirk) ---
    if (tid < NI) {
        int d        = tid;
        int m_right  = i + NI_HALF - NN * NN;
        bool left    = (i < NI_HALF) && (d < NI_HALF - i);
        bool right   = (m_right > 0) && ((d == 0) || (d >= NI - m_right));
        if (left || right) Csub[d * CS + d] = 1.0f;
    }
    // identity into right half of augmented matrix (disjoint from gram tiles)
    for (int idx = tid; idx < NI * NI; idx += 256) {
        int r = idx >> 6, c = idx & 63;
        aug[r * AS + NI + c] = (r == c) ? 1.0f : 0.0f;
    }
    __syncthreads();

    const int w    = tid >> 5;     // wave id 0..7
    const int lane = tid & 31;
    const int half = lane >> 4;    // K/M half select per ISA layout
    const int lm   = lane & 15;

    // ---- 4) gram = Csub * Csub^T  (16 tiles of 16x16, 2 per wave) ---------
    // A frag elem v: (M = tile_m + lm, K = k0 + 2*half + v)
    // B frag elem v: (K = k0 + 2*half + v, N = tile_n + lm) = Csub row tile_n+lm
    for (int t = w; t < 16; t += 8) {
        const int mt = (t >> 2) * 16;
        const int nt = (t & 3) * 16;
        v8f acc = {};
        const float* arow = &Csub[(mt + lm) * CS + 2 * half];
        const float* brow = &Csub[(nt + lm) * CS + 2 * half];
#pragma unroll
        for (int k0 = 0; k0 < NI; k0 += 4) {
            v2f a  = { arow[k0], arow[k0 + 1] };
            v2f bb = { brow[k0], brow[k0 + 1] };
            acc = __builtin_amdgcn_wmma_f32_16x16x4_f32(
                false, a, false, bb, (short)0, acc, false, false);
        }
#pragma unroll
        for (int v = 0; v < 8; ++v)
            aug[(mt + v + 8 * half) * AS + nt + lm] = acc[v];
    }
    __syncthreads();

    // ---- 5) Gauss-Jordan: aug = [gram | I] -> [I | gram^-1] ---------------
    // 4 threads per row, 32-column chunks each. SPD => no pivoting.
    const int row = tid >> 2;
    const int qc  = (tid & 3) * 32;
    for (int p = 0; p < NI; ++p) {
        float pinv = 1.0f / aug[p * AS + p];
        __syncthreads();
        if (tid < 2 * NI) aug[p * AS + tid] *= pinv;   // scale pivot row
        __syncthreads();
        if (row != p) {
            float f = aug[row * AS + p];
#pragma unroll 8
            for (int cc = 0; cc < 32; ++cc) {
                int c = qc + cc;
                aug[row * AS + c] -= f * aug[p * AS + c];
            }
        }
        __syncthreads();
    }

    // ---- 6) full_sub = inv * Csub[:, 16:48]  (8 tiles, 1 per wave) --------
    {
        const int mt = (w >> 1) * 16;
        const int nt = (w & 1) * 16;
        v8f acc = {};
        const float* arow = &aug[(mt + lm) * AS + NI + 2 * half];
#pragma unroll
        for (int k0 = 0; k0 < NI; k0 += 4) {
            v2f a  = { arow[k0], arow[k0 + 1] };
            int kb = k0 + 2 * half;
            v2f bb = { Csub[kb * CS + 16 + nt + lm],
                       Csub[(kb + 1) * CS + 16 + nt + lm] };
            acc = __builtin_amdgcn_wmma_f32_16x16x4_f32(
                false, a, false, bb, (short)0, acc, false, false);
        }
#pragma unroll
        for (int v = 0; v < 8; ++v)
            fs[(mt + v + 8 * half) * FS + nt + lm] = acc[v];
    }
    __syncthreads();

    // ---- 7) transposed, coalesced store: out[b*32+kk][j] = fs[j][kk] ------
    float* ob = out + (size_t)b * (KROW * NI);
    for (int o = tid; o < KROW * NI; o += 256)
        ob[o] = fs[(o & 63) * FS + (o >> 6)];
}

extern "C" void kernel_launch(void* const* d_in, const int* in_sizes, int n_in,
                              void* d_out, int out_size, void* d_ws, size_t ws_size,
                              hipStream_t stream) {
    (void)in_sizes; (void)n_in; (void)out_size; (void)d_ws; (void)ws_size;
    const float* Cs      = (const float*)d_in[0];
    const long long* coo = (const long long*)d_in[1];   // [2, NNZ] int64
    float* out           = (float*)d_out;               // [262144, 64] f32
    const int n_blocks   = (NN * NN) / KROW;            // 8192
    gen_cinv_rows_kernel<<<n_blocks, 256, 0, stream>>>(
        Cs, coo, coo + (size_t)NNZ, out);
}